// Codebook_35287451304792
// MI455X (gfx1250) — compile-verified
//
#include <hip/hip_runtime.h>

// VQ-VAE codebook nearest-neighbor + straight-through gather for MI455X (gfx1250).
// Dims: B=32, C=256, H=W=32 -> N = 32768 rows of D=256; K = 1024 codes.

typedef __attribute__((ext_vector_type(2))) float v2f;
typedef __attribute__((ext_vector_type(8))) float v8f;

#define CDIM    256
#define KCODES  1024
#define HW      1024            // H*W
#define BATCH   32
#define N_TOTAL (BATCH * HW)    // 32768
#define OUT_ONE ((size_t)BATCH * CDIM * HW)  // 8388608 elements per output tensor

#define ROWS_PER_BLOCK 64
#define THREADS        128      // 4 waves of 32
#define LDA            260      // padded LDS row stride (floats); 260 % 64 == 4 -> conflict-free frag reads

// ---------------------------------------------------------------------------
// Kernel 1: per-code squared norms ||e_k||^2 — one wave32 per code, coalesced
// row reads, butterfly reduce. Grid 128 x 256 threads = 1024 waves.
// ---------------------------------------------------------------------------
__global__ __launch_bounds__(256) void vq_norms_kernel(const float* __restrict__ w,
                                                       float* __restrict__ norms) {
  const int gtid = blockIdx.x * 256 + threadIdx.x;
  const int code = gtid >> 5;          // one wave per code
  const int lane = threadIdx.x & 31;
  const float* row = w + (size_t)code * CDIM;
  float s = 0.f;
#pragma unroll
  for (int c = 0; c < CDIM; c += 32) { // lane-coalesced stride-32 reads
    float v = row[c + lane];
    s = fmaf(v, v, s);
  }
#pragma unroll
  for (int m = 16; m >= 1; m >>= 1) s += __shfl_xor(s, m, 32);
  if (lane == 0) norms[code] = s;
}

// ---------------------------------------------------------------------------
// Kernel 2: argmin_k ( ||e_k||^2 - 2 * x.e_k )  via f32 WMMA 16x16x4.
// Block: 128 threads = 4 waves; 64 rows of the flattened (channels-last) z.
// A (16x4 f32 frag): lanes 0-15 hold K=0,1 ; lanes 16-31 hold K=2,3 (per ISA).
// B (4x16 f32 frag): VGPR0 = rows K=0/K=2, VGPR1 = K=1/K=3, N = lane&15.
// C/D (16x16 f32):   VGPR r = row r (lanes 0-15) / row r+8 (lanes 16-31).
// ---------------------------------------------------------------------------
__global__ __launch_bounds__(THREADS) void vq_argmin_kernel(
    const float* __restrict__ z, const float* __restrict__ w,
    const float* __restrict__ norms, int* __restrict__ out_idx) {
  __shared__ float ldsA[ROWS_PER_BLOCK * LDA];  // 66,560 B

  const int tid  = threadIdx.x;
  const int lane = tid & 31;
  const int wv   = tid >> 5;
  const int n0   = blockIdx.x * ROWS_PER_BLOCK;   // global row base (64 | 1024, so single b)
  const int b    = n0 >> 10;
  const int hw0  = n0 & (HW - 1);

  // Stage A slab: rows n0..n0+63, all 256 channels. z is [B,C,H,W]; for fixed
  // (b,c) the 64 row elements are contiguous -> coalesced 256B runs.
  for (int i = tid; i < ROWS_PER_BLOCK * CDIM; i += THREADS) {
    int c = i >> 6;                       // i / 64
    int r = i & (ROWS_PER_BLOCK - 1);     // i % 64
    ldsA[r * LDA + c] = z[(size_t)b * (CDIM * HW) + (size_t)c * HW + hw0 + r];
  }
  __syncthreads();

  const int mrow  = lane & 15;            // matrix row (A) / code-within-tile (B)
  const int khalf = (lane >> 4) << 1;     // K sub-offset: 0 for lanes 0-15, 2 for 16-31
  const float* ldsArow = &ldsA[(wv * 16 + mrow) * LDA + khalf];

  float minv[8];
  int   mini[8];
#pragma unroll
  for (int r = 0; r < 8; ++r) { minv[r] = 3.0e38f; mini[r] = 0; }

  // 16 groups of 4 code-tiles (64 codes/group) -> 4 independent WMMA acc chains.
  for (int g = 0; g < KCODES / 64; ++g) {
    const int cb = g * 64;
    const float* wp = w + (size_t)(cb + mrow) * CDIM + khalf;

    v8f acc[4] = {};

#pragma unroll 4
    for (int c = 0; c < CDIM; c += 4) {
      v2f a  = *(const v2f*)(ldsArow + c);
      v2f b0 = *(const v2f*)(wp + c);
      v2f b1 = *(const v2f*)(wp + 16 * CDIM + c);
      v2f b2 = *(const v2f*)(wp + 32 * CDIM + c);
      v2f b3 = *(const v2f*)(wp + 48 * CDIM + c);
      acc[0] = __builtin_amdgcn_wmma_f32_16x16x4_f32(false, a, false, b0, (short)0, acc[0], false, false);
      acc[1] = __builtin_amdgcn_wmma_f32_16x16x4_f32(false, a, false, b1, (short)0, acc[1], false, false);
      acc[2] = __builtin_amdgcn_wmma_f32_16x16x4_f32(false, a, false, b2, (short)0, acc[2], false, false);
      acc[3] = __builtin_amdgcn_wmma_f32_16x16x4_f32(false, a, false, b3, (short)0, acc[3], false, false);
    }

    // Fold: d = ||e||^2 - 2*S ; lane sees column n = cb + t*16 + mrow.
#pragma unroll
    for (int t = 0; t < 4; ++t) {
      const int   code = cb + t * 16 + mrow;
      const float nb   = norms[code];
#pragma unroll
      for (int r = 0; r < 8; ++r) {
        float d = fmaf(-2.f, acc[t][r], nb);
        if (d < minv[r]) { minv[r] = d; mini[r] = code; }  // strict < keeps earliest code
      }
    }
  }

  // Cross-lane reduce within each 16-lane half (rows r and r+8 live in the two
  // halves of the wave). Lexicographic (val, idx) tie-break = argmin semantics.
#pragma unroll
  for (int r = 0; r < 8; ++r) {
    float v  = minv[r];
    int   ix = mini[r];
#pragma unroll
    for (int m = 8; m >= 1; m >>= 1) {
      float vo = __shfl_xor(v, m, 32);
      int   io = __shfl_xor(ix, m, 32);
      if (vo < v || (vo == v && io < ix)) { v = vo; ix = io; }
    }
    if (mrow == 0) {
      out_idx[n0 + wv * 16 + r + ((lane >> 4) << 3)] = ix;
    }
  }
}

// ---------------------------------------------------------------------------
// Kernel 3: gather weight[idx] back to [B,C,H,W], write both outputs
// (codes and codes_bar are numerically identical in the forward pass).
// Writes are fully coalesced; gather reads hit the L2-resident 1 MB codebook.
// ---------------------------------------------------------------------------
__global__ __launch_bounds__(256) void vq_gather_kernel(
    const float* __restrict__ w, const int* __restrict__ idx,
    float* __restrict__ out) {
  size_t t = (size_t)blockIdx.x * 256 + threadIdx.x;  // index into [B,C,H,W]
  int hw = (int)(t & (HW - 1));
  int c  = (int)((t >> 10) & (CDIM - 1));
  int b  = (int)(t >> 18);
  int n  = (b << 10) | hw;
  float v = w[(size_t)idx[n] * CDIM + c];
  out[t]           = v;
  out[t + OUT_ONE] = v;
}

// ---------------------------------------------------------------------------
extern "C" void kernel_launch(void* const* d_in, const int* in_sizes, int n_in,
                              void* d_out, int out_size, void* d_ws, size_t ws_size,
                              hipStream_t stream) {
  const float* z = (const float*)d_in[0];   // [32,256,32,32]
  const float* w = (const float*)d_in[1];   // [1024,256]
  float* out = (float*)d_out;               // 2 x [32,256,32,32]

  float* norms   = (float*)d_ws;                                  // 1024 f32
  int*   indices = (int*)((char*)d_ws + KCODES * sizeof(float));  // 32768 i32

  vq_norms_kernel<<<(KCODES * 32) / 256, 256, 0, stream>>>(w, norms);
  vq_argmin_kernel<<<N_TOTAL / ROWS_PER_BLOCK, THREADS, 0, stream>>>(z, w, norms, indices);
  vq_gather_kernel<<<(int)(OUT_ONE / 256), 256, 0, stream>>>(w, indices, out);
}